// PointNet2MSG_12455405158591
// MI455X (gfx1250) — compile-verified
//
#include <hip/hip_runtime.h>
#include <stdint.h>

// Problem constants (from reference)
#define BB   16
#define CC   128
#define NN   16384
#define KK   64
#define NT   1024   // points per tile
#define TPB  256    // threads per block (8 wave32)

typedef float    v4f __attribute__((ext_vector_type(4)));
typedef unsigned v4u __attribute__((ext_vector_type(4)));

// ---- order-preserving float <-> u32 encoding (so max reduces with u32 atomics) ----
__device__ __forceinline__ unsigned enc_f32(float x) {
    unsigned b = __float_as_uint(x);
    return (b & 0x80000000u) ? ~b : (b | 0x80000000u);
}
__device__ __forceinline__ float dec_f32(unsigned u) {
    unsigned b = (u & 0x80000000u) ? (u ^ 0x80000000u) : ~u;
    return __uint_as_float(b);
}

// LDS byte offset of a __shared__ object (generic -> addrspace(3) -> u32)
__device__ __forceinline__ unsigned lds_off_u32(const void* p) {
    return (unsigned)(unsigned long long)(const __attribute__((address_space(3))) char*)p;
}

// ---- kernel 1: zero the u32 accumulator view of d_out ----
__global__ void __launch_bounds__(TPB) segmax_init(unsigned* __restrict__ out, int n) {
    int i = blockIdx.x * TPB + threadIdx.x;
    if (i < n) out[i] = 0u;   // 0 == "empty" sentinel: enc(finite) >= 0x00800000
}

// ---- kernel 2: per-tile LDS scatter-max, then global atomic merge ----
__global__ void __launch_bounds__(TPB) segmax_main(const float* __restrict__ feats,
                                                   const int*   __restrict__ labels,
                                                   unsigned*    __restrict__ out_acc) {
    __shared__ __align__(16) unsigned acc[CC * KK];  // 32 KB accumulator, layout [c][k]
    __shared__ __align__(16) int      slab[NT];      // 4 KB staged labels (async-filled)

    const int tile = blockIdx.x;       // 0..15
    const int b    = blockIdx.y;       // 0..15
    const int t    = threadIdx.x;
    const int n0   = tile * NT;

    // Kick off async label staging (CDNA5 async global->LDS, tracked by ASYNCcnt).
    // 256 lanes x 16B = 4 KB = all NT labels of this tile.
    {
        unsigned lds_dst  = lds_off_u32(&slab[0]) + (unsigned)(t * 16);
        unsigned goff     = (unsigned)((b * NN + n0 + t * 4) * (int)sizeof(int));
        unsigned long long sbase = (unsigned long long)(uintptr_t)labels;
        asm volatile("global_load_async_to_lds_b128 %0, %1, %2"
                     :: "v"(lds_dst), "v"(goff), "s"(sbase)
                     : "memory");
    }

    // Overlap: zero the LDS accumulator while the async copy is in flight.
    for (int i = t; i < CC * KK; i += TPB) acc[i] = 0u;

    asm volatile("s_wait_asynccnt 0" ::: "memory");
    __syncthreads();

    // Cache this thread's 4 labels in VGPRs (reused across all 128 channels).
    int lab0, lab1, lab2, lab3;
    {
        const int4 l4 = *reinterpret_cast<const int4*>(&slab[t * 4]);
        lab0 = l4.x; lab1 = l4.y; lab2 = l4.z; lab3 = l4.w;
    }

    // Stream the feature tile: one coalesced, non-temporal float4 per thread per channel.
    // Data is read exactly once -> TH_LOAD_NT keeps it from polluting WGP$/L2.
    const float* fbase = feats + (size_t)b * CC * NN + n0 + t * 4;
    #pragma unroll 4
    for (int c = 0; c < CC; ++c) {
        const v4f f = __builtin_nontemporal_load(
            reinterpret_cast<const v4f*>(fbase + (size_t)c * NN));
        if (c + 4 < CC)  // global_prefetch_b8 a few channels ahead
            __builtin_prefetch(fbase + (size_t)(c + 4) * NN, 0, 3);
        unsigned* row = &acc[c * KK];
        // At uniform c, LDS bank = label%64 -> distinct labels hit distinct banks.
        if (lab0 >= 0) atomicMax(&row[lab0], enc_f32(f.x));
        if (lab1 >= 0) atomicMax(&row[lab1], enc_f32(f.y));
        if (lab2 >= 0) atomicMax(&row[lab2], enc_f32(f.z));
        if (lab3 >= 0) atomicMax(&row[lab3], enc_f32(f.w));
    }
    __syncthreads();

    // Merge tile result into global accumulator (out layout [b][k][c]).
    // Vectorized LDS reads: lane reads 16B (ds_load_b128), conflict-free.
    unsigned* gout = out_acc + (size_t)b * KK * CC;
    for (int i = t * 4; i < CC * KK; i += TPB * 4) {   // i = c*KK + k, k%4 == 0
        const v4u v = *reinterpret_cast<const v4u*>(&acc[i]);
        const int c = i / KK;
        const int k = i - c * KK;
        if (v.x) atomicMax(&gout[(size_t)(k + 0) * CC + c], v.x);
        if (v.y) atomicMax(&gout[(size_t)(k + 1) * CC + c], v.y);
        if (v.z) atomicMax(&gout[(size_t)(k + 2) * CC + c], v.z);
        if (v.w) atomicMax(&gout[(size_t)(k + 3) * CC + c], v.w);
    }
}

// ---- kernel 3: decode in place; empty segments -> 0.0f ----
__global__ void __launch_bounds__(TPB) segmax_fin(float* __restrict__ out, int n) {
    int i = blockIdx.x * TPB + threadIdx.x;
    if (i < n) {
        unsigned u = __float_as_uint(out[i]);   // raw u32 accumulator bits
        out[i] = u ? dec_f32(u) : 0.0f;
    }
}

extern "C" void kernel_launch(void* const* d_in, const int* in_sizes, int n_in,
                              void* d_out, int out_size, void* d_ws, size_t ws_size,
                              hipStream_t stream) {
    (void)in_sizes; (void)n_in; (void)d_ws; (void)ws_size;
    const float* feats  = (const float*)d_in[0];   // [B, C, N] fp32
    const int*   labels = (const int*)d_in[1];     // [B, N] int32 in [-1, K)
    unsigned*    out_u  = (unsigned*)d_out;        // [B, K, C] (u32 view during accumulation)

    const int total = BB * KK * CC;                // 131072
    segmax_init<<<(total + TPB - 1) / TPB, TPB, 0, stream>>>(out_u, total);

    dim3 grid(NN / NT, BB);                        // 16 x 16 = 256 workgroups
    segmax_main<<<grid, TPB, 0, stream>>>(feats, labels, out_u);

    segmax_fin<<<(total + TPB - 1) / TPB, TPB, 0, stream>>>((float*)d_out, total);
}